// EncoderBlock_87497073754275
// MI455X (gfx1250) — compile-verified
//
#include <hip/hip_runtime.h>
#include <math.h>

// Transformer encoder block for MI455X (gfx1250), fp32 via V_WMMA_F32_16X16X4_F32.
// B=2, S=4096, D=512, H=8, Dk=Dv=64, FF=2048.

#define S_LEN   4096
#define DMODEL  512
#define NHEADS  8
#define DK      64
#define DFF     2048
#define BS_ROWS 8192   // B * S

typedef float v2f __attribute__((ext_vector_type(2)));
typedef float v8f __attribute__((ext_vector_type(8)));

__device__ __forceinline__ v8f wmma4(v2f a, v2f b, v8f c) {
  // D = A(16x4 f32) * B(4x16 f32) + C(16x16 f32)
  return __builtin_amdgcn_wmma_f32_16x16x4_f32(
      /*neg_a=*/false, a, /*neg_b=*/false, b,
      /*c_mod=*/(short)0, c, /*reuse_a=*/false, /*reuse_b=*/false);
}

__device__ __forceinline__ v8f zero8() {
  v8f z;
  for (int i = 0; i < 8; ++i) z[i] = 0.f;
  return z;
}

// A fragment: rows r0..r0+15, cols k..k+3 of row-major A (leading dim lda).
// lane&15 = row, half(lane>>4) selects K pair {k,k+1} vs {k+2,k+3}.
__device__ __forceinline__ v2f load_a_frag(const float* __restrict__ A, int lda,
                                           int r0, int k, int l16, int half) {
  const float* p = A + (size_t)(r0 + l16) * lda + (k + 2 * half);
  v2f a; a.x = p[0]; a.y = p[1];
  return a;
}

// B fragment: rows k..k+3, cols c0..c0+15 of row-major B (leading dim ldb).
// lane&15 = col, half selects K pair.
__device__ __forceinline__ v2f load_b_frag(const float* __restrict__ B, int ldb,
                                           int k, int c0, int l16, int half) {
  const float* p = B + (size_t)(k + 2 * half) * ldb + (c0 + l16);
  v2f b; b.x = p[0]; b.y = p[ldb];
  return b;
}

// reductions across the 16 lanes of a half-wave (rows of a WMMA C fragment)
__device__ __forceinline__ float rowmax16(float v) {
  v = fmaxf(v, __shfl_xor(v, 1));
  v = fmaxf(v, __shfl_xor(v, 2));
  v = fmaxf(v, __shfl_xor(v, 4));
  v = fmaxf(v, __shfl_xor(v, 8));
  return v;
}
__device__ __forceinline__ float rowsum16(float v) {
  v += __shfl_xor(v, 1);
  v += __shfl_xor(v, 2);
  v += __shfl_xor(v, 4);
  v += __shfl_xor(v, 8);
  return v;
}

// ---------------------------------------------------------------------------
// Kernel 1: QKV projections. grid=(BS_ROWS/16, 3*NHEADS), block=128 (4 waves).
// Q,V stored [bh][s][64]; K stored transposed [bh][64][s] for attention B-frags.
// ---------------------------------------------------------------------------
__global__ void __launch_bounds__(128) qkv_kernel(
    const float* __restrict__ X,
    const float* __restrict__ Wq, const float* __restrict__ Wk,
    const float* __restrict__ Wv,
    float* __restrict__ Q, float* __restrict__ Kt, float* __restrict__ V) {
  const int lane = threadIdx.x & 31, wave = threadIdx.x >> 5;
  const int l16 = lane & 15, half = lane >> 4;
  const int r0 = blockIdx.x * 16;
  const int mat = blockIdx.y >> 3, h = blockIdx.y & 7;
  const float* W = (mat == 0 ? Wq : (mat == 1 ? Wk : Wv)) + (size_t)h * DMODEL * DK;
  const int c0 = wave * 16;

  v8f c = zero8();
#pragma unroll 4
  for (int k = 0; k < DMODEL; k += 4) {
    v2f a = load_a_frag(X, DMODEL, r0, k, l16, half);
    v2f b = load_b_frag(W, DK, k, c0, l16, half);
    c = wmma4(a, b, c);
  }

  const int bidx = r0 >> 12;          // batch index
  const int s0 = r0 & (S_LEN - 1);
  if (mat == 1) {
    // K transposed: Kt[(b*8+h)*64 + col][s]
    float* base = Kt + ((size_t)(bidx * NHEADS + h) * DK + c0 + l16) * S_LEN + s0;
#pragma unroll
    for (int r = 0; r < 8; ++r) base[r + 8 * half] = c[r];
  } else {
    float* base = (mat == 0 ? Q : V) +
                  ((size_t)(bidx * NHEADS + h) * S_LEN + s0) * DK + c0 + l16;
#pragma unroll
    for (int r = 0; r < 8; ++r) base[(size_t)(r + 8 * half) * DK] = c[r];
  }
}

// ---------------------------------------------------------------------------
// Kernel 2: flash attention. grid=(S/16/8, B*H), block=256 (8 independent waves).
// Each wave: one 16-row query tile, online softmax over all 4096 keys,
// O accumulated in 4 WMMA C fragments (64 v-cols).
// ---------------------------------------------------------------------------
__global__ void __launch_bounds__(256) attn_kernel(
    const float* __restrict__ Q, const float* __restrict__ Kt,
    const float* __restrict__ V, float* __restrict__ O) {
  __shared__ float pls[8][16][20];   // P tile bounce buffer, stride 20 avoids bank conflicts
  const int lane = threadIdx.x & 31, wave = threadIdx.x >> 5;
  const int l16 = lane & 15, half = lane >> 4;
  const int rt = blockIdx.x * 8 + wave;   // row tile 0..255
  const int bh = blockIdx.y;              // 0..15
  const int s0 = rt * 16;
  const float* Qbh  = Q  + (size_t)bh * S_LEN * DK;
  const float* Ktbh = Kt + (size_t)bh * DK * S_LEN;
  const float* Vbh  = V  + (size_t)bh * S_LEN * DK;

  // Q fragments for all 16 K-steps (K=64), kept in VGPRs.
  v2f aq[16];
#pragma unroll
  for (int kb = 0; kb < 16; ++kb)
    aq[kb] = load_a_frag(Qbh, DK, s0, kb * 4, l16, half);

  v8f o0 = zero8(), o1 = zero8(), o2 = zero8(), o3 = zero8();
  float m[8], l[8];
#pragma unroll
  for (int r = 0; r < 8; ++r) { m[r] = -3.0e38f; l[r] = 0.f; }

#pragma unroll 1
  for (int t0 = 0; t0 < S_LEN; t0 += 16) {
    // scores tile S = Q * K^T  (16x16)
    v8f sc = zero8();
#pragma unroll
    for (int kb = 0; kb < 16; ++kb) {
      const float* p = Ktbh + (size_t)(kb * 4 + 2 * half) * S_LEN + t0 + l16;
      v2f bk; bk.x = p[0]; bk.y = p[S_LEN];
      sc = wmma4(aq[kb], bk, sc);
    }
    // online softmax update per row (row = r + 8*half, striped over 16 lanes)
#pragma unroll
    for (int r = 0; r < 8; ++r) {
      float v = sc[r] * 0.125f;                 // 1/sqrt(64)
      float mnew = fmaxf(m[r], rowmax16(v));
      float alpha = __expf(m[r] - mnew);
      float pe = __expf(v - mnew);
      l[r] = l[r] * alpha + rowsum16(pe);
      m[r] = mnew;
      o0[r] *= alpha; o1[r] *= alpha; o2[r] *= alpha; o3[r] *= alpha;
      pls[wave][r + 8 * half][l16] = pe;        // C-layout -> LDS
    }
    // O += P * V   (A-frags of P read back from LDS in A-layout)
#pragma unroll
    for (int kb = 0; kb < 16; kb += 4) {
      v2f ap;
      ap.x = pls[wave][l16][kb + 2 * half];
      ap.y = pls[wave][l16][kb + 2 * half + 1];
      const float* vp = Vbh + (size_t)(t0 + kb + 2 * half) * DK + l16;
      v2f b0;  b0.x  = vp[0];  b0.y  = vp[DK];
      v2f b1v; b1v.x = vp[16]; b1v.y = vp[DK + 16];
      v2f b2v; b2v.x = vp[32]; b2v.y = vp[DK + 32];
      v2f b3v; b3v.x = vp[48]; b3v.y = vp[DK + 48];
      o0 = wmma4(ap, b0,  o0);
      o1 = wmma4(ap, b1v, o1);
      o2 = wmma4(ap, b2v, o2);
      o3 = wmma4(ap, b3v, o3);
    }
  }

  const int b = bh >> 3, h = bh & 7;
#pragma unroll
  for (int r = 0; r < 8; ++r) {
    float inv = 1.0f / l[r];
    size_t row = (size_t)b * S_LEN + s0 + r + 8 * half;
    float* op = O + row * (NHEADS * DK) + h * DK + l16;
    op[0]  = o0[r] * inv;
    op[16] = o1[r] * inv;
    op[32] = o2[r] * inv;
    op[48] = o3[r] * inv;
  }
}

// ---------------------------------------------------------------------------
// Kernel 3/5: GEMM (A[8192 x kdim] * W[kdim x 512]) + bias? + residual + LayerNorm.
// One block owns 16 complete rows: 8 waves x 4 col-tiles = 512 cols. grid=512, block=256.
// ---------------------------------------------------------------------------
__global__ void __launch_bounds__(256) gemm_ln_kernel(
    const float* __restrict__ A, int lda, int kdim,
    const float* __restrict__ W, const float* __restrict__ bias,
    const float* __restrict__ resid,
    const float* __restrict__ g, const float* __restrict__ bta,
    float* __restrict__ out) {
  __shared__ float rowbuf[16][DMODEL + 4];
  const int lane = threadIdx.x & 31, wave = threadIdx.x >> 5;
  const int l16 = lane & 15, half = lane >> 4;
  const int r0 = blockIdx.x * 16;
  const int c0 = wave * 64;

  v8f c[4];
#pragma unroll
  for (int j = 0; j < 4; ++j) c[j] = zero8();

#pragma unroll 4
  for (int k = 0; k < kdim; k += 4) {
    v2f a = load_a_frag(A, lda, r0, k, l16, half);
    __builtin_prefetch(A + (size_t)(r0 + l16) * lda + k + 64, 0, 1);
#pragma unroll
    for (int j = 0; j < 4; ++j) {
      v2f b = load_b_frag(W, DMODEL, k, c0 + j * 16, l16, half);
      c[j] = wmma4(a, b, c[j]);
    }
  }

  // epilogue: bias + residual -> LDS row buffer
#pragma unroll
  for (int j = 0; j < 4; ++j) {
    int col = c0 + j * 16 + l16;
    float bv = bias ? bias[col] : 0.f;
#pragma unroll
    for (int r = 0; r < 8; ++r) {
      int row = r + 8 * half;
      rowbuf[row][col] = c[j][r] + bv + resid[(size_t)(r0 + row) * DMODEL + col];
    }
  }
  __syncthreads();

  // LayerNorm: 16 threads per row, 32 elems each
  const int row = threadIdx.x >> 4;
  const int i0 = threadIdx.x & 15;
  float sum = 0.f, sq = 0.f;
#pragma unroll
  for (int jj = 0; jj < 32; ++jj) {
    float v = rowbuf[row][i0 + 16 * jj];
    sum += v; sq += v * v;
  }
  sum = rowsum16(sum); sq = rowsum16(sq);
  float mu = sum * (1.0f / 512.0f);
  float var = sq * (1.0f / 512.0f) - mu * mu;
  float rs = rsqrtf(var + 1e-5f);
#pragma unroll
  for (int jj = 0; jj < 32; ++jj) {
    int col = i0 + 16 * jj;
    float v = rowbuf[row][col];
    out[(size_t)(r0 + row) * DMODEL + col] = (v - mu) * rs * g[col] + bta[col];
  }
}

// ---------------------------------------------------------------------------
// Kernel 4: FFN1 = relu(X1 * W1 + b1). grid=(512,16), block=256; one wave per 16-col tile.
// ---------------------------------------------------------------------------
__global__ void __launch_bounds__(256) ffn1_kernel(
    const float* __restrict__ X1, const float* __restrict__ W1,
    const float* __restrict__ b1, float* __restrict__ H1) {
  const int lane = threadIdx.x & 31, wave = threadIdx.x >> 5;
  const int l16 = lane & 15, half = lane >> 4;
  const int r0 = blockIdx.x * 16;
  const int c0 = (blockIdx.y * 8 + wave) * 16;

  v8f c = zero8();
#pragma unroll 4
  for (int k = 0; k < DMODEL; k += 4) {
    v2f a = load_a_frag(X1, DMODEL, r0, k, l16, half);
    v2f b = load_b_frag(W1, DFF, k, c0, l16, half);
    c = wmma4(a, b, c);
  }
  float bv = b1[c0 + l16];
#pragma unroll
  for (int r = 0; r < 8; ++r) {
    float v = fmaxf(c[r] + bv, 0.f);
    H1[(size_t)(r0 + r + 8 * half) * DFF + c0 + l16] = v;
  }
}

// ---------------------------------------------------------------------------
extern "C" void kernel_launch(void* const* d_in, const int* in_sizes, int n_in,
                              void* d_out, int out_size, void* d_ws, size_t ws_size,
                              hipStream_t stream) {
  (void)in_sizes; (void)n_in; (void)out_size; (void)ws_size;
  const float* x    = (const float*)d_in[0];
  const float* Wq   = (const float*)d_in[1];
  const float* Wk   = (const float*)d_in[2];
  const float* Wv   = (const float*)d_in[3];
  const float* Wo   = (const float*)d_in[4];
  const float* ln1g = (const float*)d_in[5];
  const float* ln1b = (const float*)d_in[6];
  const float* W1   = (const float*)d_in[7];
  const float* b1   = (const float*)d_in[8];
  const float* W2   = (const float*)d_in[9];
  const float* b2   = (const float*)d_in[10];
  const float* ln2g = (const float*)d_in[11];
  const float* ln2b = (const float*)d_in[12];

  float* ws = (float*)d_ws;
  const size_t QN = (size_t)2 * NHEADS * S_LEN * DK;  // 4,194,304 floats
  float* Q  = ws;
  float* Kt = ws + QN;
  float* V  = ws + 2 * QN;
  float* O  = ws + 3 * QN;
  float* X1 = ws;            // reuses Q region (dead after attention)
  float* H1 = ws + QN;       // reuses Kt/V/O region (dead after out-proj)

  qkv_kernel<<<dim3(BS_ROWS / 16, 3 * NHEADS), 128, 0, stream>>>(x, Wq, Wk, Wv, Q, Kt, V);
  attn_kernel<<<dim3(S_LEN / 16 / 8, 2 * NHEADS), 256, 0, stream>>>(Q, Kt, V, O);
  gemm_ln_kernel<<<BS_ROWS / 16, 256, 0, stream>>>(O, DMODEL, DMODEL, Wo, nullptr,
                                                   x, ln1g, ln1b, X1);
  ffn1_kernel<<<dim3(BS_ROWS / 16, DFF / 128), 256, 0, stream>>>(X1, W1, b1, H1);
  gemm_ln_kernel<<<BS_ROWS / 16, 256, 0, stream>>>(H1, DFF, DFF, W2, b2,
                                                   X1, ln2g, ln2b, (float*)d_out);
}